// PR3d_11046655885449
// MI455X (gfx1250) — compile-verified
//
#include <hip/hip_runtime.h>

// ---------------------------------------------------------------------------
// Pole-residue 3D spectral operator for MI455X (gfx1250, wave32).
//  alpha = DFT3(x); or1[f] = alpha[f](16x16) @ H[f](16x16) complex GEMM via
//  V_WMMA_F32_16X16X4_F32; x1 = Re(iDFT3(or1))/F; x2 path via factorized
//  rank-6 contractions. All scratch lives in d_ws (~88 MB, L2 resident).
// ---------------------------------------------------------------------------

constexpr int BB = 16, CICO = 256, MM = 6;
constexpr int NT = 39, NX = 14, NY = 14;
constexpr int FT = NT * NX * NY;     // 7644 frequencies
constexpr int BC = 256;              // B*CI
constexpr float TWO_PI = 6.28318530717958647692f;

typedef float v2f __attribute__((ext_vector_type(2)));
typedef float v8f __attribute__((ext_vector_type(8)));

__device__ __forceinline__ float2 cfma(float2 a, float2 b, float2 c) {
  c.x = fmaf(a.x, b.x, fmaf(-a.y, b.y, c.x));
  c.y = fmaf(a.x, b.y, fmaf(a.y, b.x, c.y));
  return c;
}

// ---------------- DFT passes (matrix DFT, VALU + transcendental) -----------

// Forward pass 1: real input, Y axis (N=14, stride=1).  x -> c0
__global__ void dft_fwd_first(const float* __restrict__ x, float2* __restrict__ out) {
  int e = blockIdx.x * blockDim.x + threadIdx.x;
  if (e >= BC * FT) return;
  int ky = e % NY;
  const float* base = x + (size_t)(e / NY) * NY;
  float2 acc = {0.f, 0.f};
  for (int j = 0; j < NY; ++j) {
    float ang = -TWO_PI * (float)((ky * j) % NY) / (float)NY;
    float s, c; __sincosf(ang, &s, &c);
    float v = base[j];
    acc.x = fmaf(c, v, acc.x);
    acc.y = fmaf(s, v, acc.y);
  }
  out[e] = acc;
}

// Generic complex DFT pass along an axis of length N with element stride.
__global__ void dft_pass_c(const float2* __restrict__ in, float2* __restrict__ out,
                           int N, int stride, float sign) {
  int e = blockIdx.x * blockDim.x + threadIdx.x;
  if (e >= BC * FT) return;
  int inner = e % stride;
  int k = (e / stride) % N;
  int outer = e / (stride * N);
  const float2* base = in + (size_t)outer * N * stride + inner;
  float2 acc = {0.f, 0.f};
  for (int j = 0; j < N; ++j) {
    float ang = sign * TWO_PI * (float)((k * j) % N) / (float)N;
    float s, c; __sincosf(ang, &s, &c);
    float2 v = base[(size_t)j * stride];
    acc.x += c * v.x - s * v.y;
    acc.y += c * v.y + s * v.x;
  }
  out[e] = acc;
}

// Inverse final pass: T axis (N=39, stride=196), real output, scaled.
__global__ void dft_inv_last_real(const float2* __restrict__ in, float* __restrict__ out,
                                  float scale) {
  int e = blockIdx.x * blockDim.x + threadIdx.x;
  if (e >= BC * FT) return;
  int inner = e % (NX * NY);
  int k = (e / (NX * NY)) % NT;
  int outer = e / FT;
  const float2* base = in + (size_t)outer * FT + inner;
  float acc = 0.f;
  for (int j = 0; j < NT; ++j) {
    float ang = TWO_PI * (float)((k * j) % NT) / (float)NT;
    float s, c; __sincosf(ang, &s, &c);
    float2 v = base[(size_t)j * (NX * NY)];
    acc += c * v.x - s * v.y;
  }
  out[e] = acc * scale;
}

// --------------- pole reciprocals r = 1/(i*w - p) and exponentials ---------

// r_out: [N][ik][M],  e_out: [ik][M][N].  lam = 2*pi*i*fftfreq(N,d), t = n*d.
__global__ void precompute_axis(const float* __restrict__ wp_re, const float* __restrict__ wp_im,
                                float2* __restrict__ r_out, float2* __restrict__ e_out,
                                int N, float d) {
  int t = blockIdx.x * blockDim.x + threadIdx.x;
  int total = N * CICO * MM;
  if (t >= total) return;
  int n = t / (CICO * MM);
  int rem = t % (CICO * MM);
  int ik = rem / MM;
  int p = rem % MM;
  float pre = wp_re[ik * MM + p];
  float pim = wp_im[ik * MM + p];
  int fw = (n <= (N - 1) / 2) ? n : n - N;
  float w = TWO_PI * (float)fw / ((float)N * d);
  // r = 1/(-pre + i*(w - pim))
  float dr = -pre, di = w - pim;
  float inv = 1.0f / (dr * dr + di * di);
  r_out[(n * CICO + ik) * MM + p] = make_float2(dr * inv, -di * inv);
  // e = exp((pre + i*pim) * tval)
  float tv = (float)n * d;
  float mag = __expf(pre * tv);
  float s, c; __sincosf(pim * tv, &s, &c);
  e_out[(ik * MM + p) * N + n] = make_float2(mag * c, mag * s);
}

// --------------- H[i,k,m,n,l] = sum_pqr wr*r1*r2*r3  (per (i,k) WG) --------

__global__ void build_H(const float* __restrict__ wr_re, const float* __restrict__ wr_im,
                        const float2* __restrict__ r1, const float2* __restrict__ r2,
                        const float2* __restrict__ r3,
                        float* __restrict__ Hre, float* __restrict__ Him) {
  int ik = blockIdx.x;            // 0..255
  __shared__ float2 wr_s[216];
  __shared__ float2 r1_s[NT * MM];
  __shared__ float2 r2_s[NX * MM];
  __shared__ float2 r3_s[NY * MM];
  __shared__ float2 A_s[MM * MM * NY];     // [p*6+q][l]
  __shared__ float2 Bt_s[MM * NX * NY];    // [p][n*14+l]
  int t = threadIdx.x;
  for (int u = t; u < 216; u += 256)
    wr_s[u] = make_float2(wr_re[ik * 216 + u], wr_im[ik * 216 + u]);
  for (int u = t; u < NT * MM; u += 256)
    r1_s[u] = r1[((u / MM) * CICO + ik) * MM + (u % MM)];
  for (int u = t; u < NX * MM; u += 256) {
    r2_s[u] = r2[((u / MM) * CICO + ik) * MM + (u % MM)];
    r3_s[u] = r3[((u / MM) * CICO + ik) * MM + (u % MM)];
  }
  __syncthreads();
  for (int u = t; u < MM * MM * NY; u += 256) {   // A[pq][l] = sum_r wr*r3
    int l = u % NY, pq = u / NY;
    float2 acc = {0.f, 0.f};
    for (int r = 0; r < MM; ++r) acc = cfma(wr_s[pq * MM + r], r3_s[l * MM + r], acc);
    A_s[pq * NY + l] = acc;
  }
  __syncthreads();
  for (int u = t; u < MM * NX * NY; u += 256) {   // Bt[p][n][l] = sum_q A*r2
    int l = u % NY, n = (u / NY) % NX, p = u / (NX * NY);
    float2 acc = {0.f, 0.f};
    for (int q = 0; q < MM; ++q) acc = cfma(A_s[(p * MM + q) * NY + l], r2_s[n * MM + q], acc);
    Bt_s[(p * NX + n) * NY + l] = acc;
  }
  __syncthreads();
  for (int u = t; u < FT; u += 256) {             // H[m][nl] = sum_p Bt*r1
    int nl = u % (NX * NY), m = u / (NX * NY);
    float2 acc = {0.f, 0.f};
    for (int p = 0; p < MM; ++p) acc = cfma(Bt_s[p * NX * NY + nl], r1_s[m * MM + p], acc);
    Hre[(size_t)u * 256 + ik] = acc.x;
    Him[(size_t)u * 256 + ik] = acc.y;
  }
}

// --------------- repack alpha [bi][f] -> planar [f][bi] --------------------

__global__ void repack_alpha(const float2* __restrict__ a, float* __restrict__ are,
                             float* __restrict__ aim) {
  int e = blockIdx.x * blockDim.x + threadIdx.x;
  if (e >= BC * FT) return;
  int bc = e / FT, f = e % FT;
  float2 v = a[e];
  are[(size_t)f * 256 + bc] = v.x;
  aim[(size_t)f * 256 + bc] = v.y;
}

// --------------- or1: 7644 x (16x16x16 complex GEMM) via f32 WMMA ----------
// One wave per frequency. D = A(alpha, MxK=b x i) * B(H, KxN=i x k).
// Cre = Are*Bre - Aim*Bim ; Cim = Are*Bim + Aim*Bre  (4 real WMMA chains).

__global__ void wmma_or1(const float* __restrict__ Are, const float* __restrict__ Aim,
                         const float* __restrict__ Hre, const float* __restrict__ Him,
                         float2* __restrict__ or1c) {
  int wave = threadIdx.x >> 5;
  int lane = threadIdx.x & 31;
  int f = blockIdx.x * 8 + wave;
  if (f >= FT) return;                    // uniform per wave -> EXEC stays full
  const float* ar = Are + (size_t)f * 256;
  const float* ai = Aim + (size_t)f * 256;
  const float* hr = Hre + (size_t)f * 256;
  const float* hi = Him + (size_t)f * 256;
  int hi16 = (lane < 16) ? 0 : 1;
  int row = lane & 15;                    // A row (b) / B,N col (k)
  v8f prr = {0.f,0.f,0.f,0.f,0.f,0.f,0.f,0.f};
  v8f pii = {0.f,0.f,0.f,0.f,0.f,0.f,0.f,0.f};
  v8f pri = {0.f,0.f,0.f,0.f,0.f,0.f,0.f,0.f};
  v8f pir = {0.f,0.f,0.f,0.f,0.f,0.f,0.f,0.f};
#pragma unroll
  for (int kk = 0; kk < 4; ++kk) {
    int i0 = kk * 4 + hi16 * 2;           // K pair handled by this lane half
    float2 arp = *(const float2*)(ar + row * 16 + i0);
    float2 aip = *(const float2*)(ai + row * 16 + i0);
    v2f a_re = {arp.x, arp.y};
    v2f a_im = {aip.x, aip.y};
    v2f b_re = {hr[i0 * 16 + row], hr[(i0 + 1) * 16 + row]};
    v2f b_im = {hi[i0 * 16 + row], hi[(i0 + 1) * 16 + row]};
    prr = __builtin_amdgcn_wmma_f32_16x16x4_f32(false, a_re, false, b_re, (short)0, prr, false, false);
    pii = __builtin_amdgcn_wmma_f32_16x16x4_f32(false, a_im, false, b_im, (short)0, pii, false, false);
    pri = __builtin_amdgcn_wmma_f32_16x16x4_f32(false, a_re, false, b_im, (short)0, pri, false, false);
    pir = __builtin_amdgcn_wmma_f32_16x16x4_f32(false, a_im, false, b_re, (short)0, pir, false, false);
  }
#pragma unroll
  for (int j = 0; j < 8; ++j) {
    int brow = j + hi16 * 8;              // D row = batch b, col = co k
    float2 v = make_float2(prr[j] - pii[j], pri[j] + pir[j]);
    or1c[(size_t)(brow * 16 + row) * FT + f] = v;
  }
}

// --------------- S[b,i,k,p,q,r] = sum_mnl alpha*r1*r2*r3  (per (b,i,k) WG) -

__global__ void build_S(const float2* __restrict__ alpha,  // [bi][f]
                        const float2* __restrict__ r1, const float2* __restrict__ r2,
                        const float2* __restrict__ r3, float2* __restrict__ S) {
  int wg = blockIdx.x;                  // (b*16+i)*16+k
  int k = wg % 16;
  int bi = wg / 16;
  int ik = (bi % 16) * 16 + k;
  __shared__ float2 r1_s[NT * MM];
  __shared__ float2 r2_s[NX * MM];
  __shared__ float2 r3_s[NY * MM];
  __shared__ float2 t1[MM * NX * NY];   // [p][nl]
  __shared__ float2 t2[MM * MM * NY];   // [pq][l]
  int t = threadIdx.x;
  for (int u = t; u < NT * MM; u += 256)
    r1_s[u] = r1[((u / MM) * CICO + ik) * MM + (u % MM)];
  for (int u = t; u < NX * MM; u += 256) {
    r2_s[u] = r2[((u / MM) * CICO + ik) * MM + (u % MM)];
    r3_s[u] = r3[((u / MM) * CICO + ik) * MM + (u % MM)];
  }
  __syncthreads();
  if (t < NX * NY) {                    // t1[p][nl] = sum_m alpha[m][nl]*r1[m][p]
    float2 acc[MM];
    for (int p = 0; p < MM; ++p) acc[p] = make_float2(0.f, 0.f);
    for (int m = 0; m < NT; ++m) {
      float2 av = alpha[(size_t)bi * FT + m * (NX * NY) + t];
      for (int p = 0; p < MM; ++p) acc[p] = cfma(av, r1_s[m * MM + p], acc[p]);
    }
    for (int p = 0; p < MM; ++p) t1[p * NX * NY + t] = acc[p];
  }
  __syncthreads();
  for (int u = t; u < MM * MM * NY; u += 256) {  // t2[pq][l] = sum_n t1*r2
    int l = u % NY, q = (u / NY) % MM, p = u / (MM * NY);
    float2 acc = {0.f, 0.f};
    for (int n = 0; n < NX; ++n) acc = cfma(t1[(p * NX + n) * NY + l], r2_s[n * MM + q], acc);
    t2[(p * MM + q) * NY + l] = acc;
  }
  __syncthreads();
  for (int u = t; u < 216; u += 256) {           // S[pqr] = sum_l t2*r3
    int r = u % MM, pq = u / MM;
    float2 acc = {0.f, 0.f};
    for (int l = 0; l < NY; ++l) acc = cfma(t2[pq * NY + l], r3_s[l * MM + r], acc);
    S[(size_t)wg * 216 + u] = acc;
  }
}

// --------------- or2[b,k,pqr] = -sum_i wr[i,k,pqr]*S[b,i,k,pqr] ------------

__global__ void build_or2(const float* __restrict__ wr_re, const float* __restrict__ wr_im,
                          const float2* __restrict__ S, float2* __restrict__ or2) {
  int e = blockIdx.x * blockDim.x + threadIdx.x;
  if (e >= BB * 16 * 216) return;
  int pqr = e % 216;
  int bk = e / 216;
  int k = bk % 16, b = bk / 16;
  float2 acc = {0.f, 0.f};
  for (int i = 0; i < 16; ++i) {
    float2 w = make_float2(wr_re[(i * 16 + k) * 216 + pqr], wr_im[(i * 16 + k) * 216 + pqr]);
    acc = cfma(w, S[((size_t)(b * 16 + i) * 16 + k) * 216 + pqr], acc);
  }
  or2[e] = make_float2(-acc.x, -acc.y);
}

// --------------- x2 path, fused per (batch, co), adds into d_out -----------

__global__ void x2_add(const float2* __restrict__ or2,
                       const float2* __restrict__ e1, const float2* __restrict__ e2,
                       const float2* __restrict__ e3, float* __restrict__ out) {
  int wg = blockIdx.x;                  // b*16 + co
  int co = wg % 16, b = wg / 16;
  __shared__ float2 o2s[16 * 216];      // [c'][pqr]
  __shared__ float2 U[MM * MM * NY];    // [pq][y]
  __shared__ float2 V[MM * NX * NY];    // [p][x*14+y]
  int t = threadIdx.x;
  for (int u = t; u < 16 * 216; u += 256) o2s[u] = or2[(size_t)b * 16 * 216 + u];
  float acc[30];
  for (int j = 0; j < 30; ++j) acc[j] = 0.f;
  __syncthreads();
  for (int c = 0; c < 16; ++c) {
    for (int u = t; u < MM * MM * NY; u += 256) {   // U = sum_r or2*e3
      int y = u % NY, pq = u / NY;
      float2 a = {0.f, 0.f};
      for (int r = 0; r < MM; ++r)
        a = cfma(o2s[c * 216 + pq * MM + r], e3[((c * 16 + co) * MM + r) * NY + y], a);
      U[pq * NY + y] = a;
    }
    __syncthreads();
    for (int u = t; u < MM * NX * NY; u += 256) {   // V = sum_q U*e2
      int y = u % NY, x = (u / NY) % NX, p = u / (NX * NY);
      float2 a = {0.f, 0.f};
      for (int q = 0; q < MM; ++q)
        a = cfma(U[(p * MM + q) * NY + y], e2[((c * 16 + co) * MM + q) * NX + x], a);
      V[(p * NX + x) * NY + y] = a;
    }
    __syncthreads();
    for (int j = 0, u = t; u < FT; ++j, u += 256) { // Re(sum_p V*e1)
      int xy = u % (NX * NY), z = u / (NX * NY);
      float a = 0.f;
      for (int p = 0; p < MM; ++p) {
        float2 v = V[p * NX * NY + xy];
        float2 ee = e1[((c * 16 + co) * MM + p) * NT + z];
        a += v.x * ee.x - v.y * ee.y;
      }
      acc[j] += a;
    }
    __syncthreads();
  }
  const float sc = 1.0f / (float)FT;
  for (int j = 0, u = t; u < FT; ++j, u += 256)
    out[(size_t)wg * FT + u] += acc[j] * sc;
}

// ---------------------------------------------------------------------------

extern "C" void kernel_launch(void* const* d_in, const int* in_sizes, int n_in,
                              void* d_out, int out_size, void* d_ws, size_t ws_size,
                              hipStream_t stream) {
  (void)in_sizes; (void)n_in; (void)out_size; (void)ws_size;
  const float* x      = (const float*)d_in[0];
  const float* wp1_re = (const float*)d_in[1];
  const float* wp1_im = (const float*)d_in[2];
  const float* wp2_re = (const float*)d_in[3];
  const float* wp2_im = (const float*)d_in[4];
  const float* wp3_re = (const float*)d_in[5];
  const float* wp3_im = (const float*)d_in[6];
  const float* wr_re  = (const float*)d_in[7];
  const float* wr_im  = (const float*)d_in[8];
  float* out = (float*)d_out;

  float* w = (float*)d_ws;
  size_t o = 0;
  auto A2 = [&](size_t n) { float2* p = (float2*)(w + o); o += 2 * n; return p; };
  auto A1 = [&](size_t n) { float* p = w + o; o += n; return p; };
  float2* c0   = A2((size_t)BC * FT);
  float2* c1   = A2((size_t)BC * FT);
  float*  Are  = A1((size_t)BC * FT);
  float*  Aim  = A1((size_t)BC * FT);
  float*  Hre  = A1((size_t)BC * FT);
  float*  Him  = A1((size_t)BC * FT);
  float2* or1c = A2((size_t)BC * FT);
  float2* r1   = A2((size_t)NT * CICO * MM);
  float2* r2   = A2((size_t)NX * CICO * MM);
  float2* r3   = A2((size_t)NY * CICO * MM);
  float2* e1   = A2((size_t)CICO * MM * NT);
  float2* e2   = A2((size_t)CICO * MM * NX);
  float2* e3   = A2((size_t)CICO * MM * NY);
  float2* Sb   = A2((size_t)BB * CICO * 216);
  float2* or2  = A2((size_t)BB * 16 * 216);

  dim3 blk(256);
  int gE = (BC * FT + 255) / 256;   // 7644 blocks

  precompute_axis<<<dim3((NT * CICO * MM + 255) / 256), blk, 0, stream>>>(wp1_re, wp1_im, r1, e1, NT, 1.0f);
  precompute_axis<<<dim3((NX * CICO * MM + 255) / 256), blk, 0, stream>>>(wp2_re, wp2_im, r2, e2, NX, 1.0f / 27.0f);
  precompute_axis<<<dim3((NY * CICO * MM + 255) / 256), blk, 0, stream>>>(wp3_re, wp3_im, r3, e3, NY, 1.0f / 27.0f);

  // Forward DFT: Y, X, T  ->  alpha in c0  (layout [b*16+i][t*196+x*14+y])
  dft_fwd_first<<<gE, blk, 0, stream>>>(x, c0);
  dft_pass_c<<<gE, blk, 0, stream>>>(c0, c1, NX, NX, -1.0f);
  dft_pass_c<<<gE, blk, 0, stream>>>(c1, c0, NT, NX * NY, -1.0f);

  // Per-frequency complex GEMM on the matrix pipe
  repack_alpha<<<gE, blk, 0, stream>>>(c0, Are, Aim);
  build_H<<<dim3(CICO), blk, 0, stream>>>(wr_re, wr_im, r1, r2, r3, Hre, Him);
  wmma_or1<<<dim3((FT + 7) / 8), blk, 0, stream>>>(Are, Aim, Hre, Him, or1c);

  // or2 path (needs alpha in c0 before inverse DFT clobbers it)
  build_S<<<dim3(BB * CICO), blk, 0, stream>>>(c0, r1, r2, r3, Sb);
  build_or2<<<dim3((BB * 16 * 216 + 255) / 256), blk, 0, stream>>>(wr_re, wr_im, Sb, or2);

  // x1 = Re(iDFT3(or1))/F  written straight into d_out
  dft_pass_c<<<gE, blk, 0, stream>>>(or1c, c0, NY, 1, 1.0f);
  dft_pass_c<<<gE, blk, 0, stream>>>(c0, c1, NX, NX, 1.0f);
  dft_inv_last_real<<<gE, blk, 0, stream>>>(c1, out, 1.0f / (float)FT);

  // x2 accumulated on top
  x2_add<<<dim3(BB * 16), blk, 0, stream>>>(or2, e1, e2, e3, out);
}